// LearnableDynamicalOperator_49890340111075
// MI455X (gfx1250) — compile-verified
//
#include <hip/hip_runtime.h>
#include <hip/hip_bf16.h>

// dynoNet G-block on gfx1250 via chunked-convolution WMMA.
// y_io[t] = sum_k b_iok u_i[t-k] - sum_k a_iok y_io[t-1-k];  out[b,t,o] = sum_i y_io[t]
//
// Chunked form (L=64):  Y_chunk[64 x 16b] = M_io[64 x 72] * X[72 x 16b]
//   X rows 0..63  = u[t0 + s]          (s = 0..63)
//   X row  64     = u[t0 - 1]
//   X row  65     = u[t0 - 2]
//   X rows 66..68 = y[t0-1], y[t0-2], y[t0-3]   (carry, fed back per chunk)
//   X rows 69..71 = 0 (pad so K % 4 == 0 for V_WMMA_F32_16X16X4_F32)
// M columns: hfull (lower-triangular conv), p1/p2 (FIR history), g0..g2 (IIR carry).

typedef __attribute__((ext_vector_type(2))) float v2f;
typedef __attribute__((ext_vector_type(8))) float v8f;

#define Bc 128
#define Tc 4096
#define Ic 8
#define Oc 8
#define Lc 64      // chunk length (time)
#define KP 72      // padded K (64 + 2 + 3 -> 72)
#define BT 16      // batch tile (WMMA N)

// ---------------------------------------------------------------------------
// Kernel 1: per-(i,o) impulse responses -> transposed matrix Mt[pair][k][m]
// ---------------------------------------------------------------------------
__global__ void dyno_precompute(const float* __restrict__ num,
                                const float* __restrict__ den,
                                float* __restrict__ Mt)
{
    const int p = threadIdx.x;          // pair = i*Oc + o, 64 threads
    if (p >= Ic * Oc) return;
    const float b0 = num[p * 3 + 0], b1 = num[p * 3 + 1], b2 = num[p * 3 + 2];
    const float a0 = den[p * 3 + 0], a1 = den[p * 3 + 1], a2 = den[p * 3 + 2];

    // h = impulse response of 1/a(z): h[0]=1, h[t] = -(a0 h[t-1]+a1 h[t-2]+a2 h[t-3])
    float h[Lc];
    h[0] = 1.0f;
    for (int t = 1; t < Lc; ++t) {
        float v = -a0 * h[t - 1];
        if (t >= 2) v -= a1 * h[t - 2];
        if (t >= 3) v -= a2 * h[t - 3];
        h[t] = v;
    }

    float* Mp = Mt + (size_t)p * KP * Lc;   // Mt[pair][k][m] = M[m][k]
    for (int k = 0; k < KP; ++k) {
        for (int m = 0; m < Lc; ++m) {
            float v = 0.0f;
            if (k < Lc) {                        // hfull[m-k] = (b conv h)[m-k]
                const int d = m - k;
                if (d >= 0) {
                    v = b0 * h[d];
                    if (d >= 1) v += b1 * h[d - 1];
                    if (d >= 2) v += b2 * h[d - 2];
                }
            } else if (k == 64) {                // response to u[-1]
                v = b1 * h[m] + (m >= 1 ? b2 * h[m - 1] : 0.0f);
            } else if (k == 65) {                // response to u[-2]
                v = b2 * h[m];
            } else if (k == 66) {                // response to y[-1]
                v = -(a0 * h[m] + (m >= 1 ? a1 * h[m - 1] : 0.0f)
                                + (m >= 2 ? a2 * h[m - 2] : 0.0f));
            } else if (k == 67) {                // response to y[-2]
                v = -(a1 * h[m] + (m >= 1 ? a2 * h[m - 1] : 0.0f));
            } else if (k == 68) {                // response to y[-3]
                v = -a2 * h[m];
            }                                    // k = 69..71: pad zeros
            Mp[k * Lc + m] = v;
        }
    }
}

// ---------------------------------------------------------------------------
// Kernel 2: 64 sequential chunk-GEMMs with WMMA f32 16x16x4
// grid = Oc * (Bc/BT) = 64 blocks, 256 threads = 8 waves
// wave w: tile = w&3 (16 time rows), ihalf = w>>2 (4 of 8 input channels)
// ---------------------------------------------------------------------------
__device__ __forceinline__ float read_x(const float* __restrict__ u_lds,
                                        float (*cbuf)[3][16],
                                        int k, int n, int i)
{
    if (k < 64)  return u_lds[((k + 2) * 16 + n) * 9 + i];   // u[t0+k] at tl=k+2
    if (k == 64) return u_lds[((1) * 16 + n) * 9 + i];       // u[t0-1]
    if (k == 65) return u_lds[((0) * 16 + n) * 9 + i];       // u[t0-2]
    if (k < 69)  return cbuf[i][k - 66][n];                  // y[t0-1-(k-66)]
    return 0.0f;                                             // pad
}

__global__ __launch_bounds__(256, 1)
void dyno_main(const float* __restrict__ u,
               const float* __restrict__ Mt,
               float* __restrict__ out)
{
    // LDS: U chunk (pad 8->9 floats per (t,b) row: coalesced load, conflict-free read)
    __shared__ float u_lds[66 * 16 * 9];          // 38016 B
    __shared__ float carry[2][Ic][3][16];         // double-buffered y carry, 3072 B
    __shared__ float red[4][16][16];              // cross-i-half reduction, 4096 B

    const int tid   = threadIdx.x;
    const int lane  = tid & 31;                   // wave32
    const int w     = tid >> 5;
    const int tile  = w & 3;                      // time tile 0..3 (rows tile*16..+15)
    const int ihalf = w >> 2;                     // 0: i=0..3, 1: i=4..7
    const int n     = lane & 15;                  // WMMA N (batch within tile)
    const int hi    = lane >> 4;                  // lane half -> K pair select

    const int nbt = Bc / BT;                      // 8
    const int o   = blockIdx.x / nbt;
    const int b0  = (blockIdx.x % nbt) * BT;

    for (int idx = tid; idx < 2 * Ic * 3 * 16; idx += 256)
        ((float*)carry)[idx] = 0.0f;              // y[<0] = 0

    for (int c = 0; c < Tc / Lc; ++c) {
        const int t0 = c * Lc;
        __syncthreads();                          // prev chunk fully consumed LDS

        // Stage u[b0..b0+15, t0-2 .. t0+63, 0..7] into LDS (coalesced 32B groups)
        for (int idx = tid; idx < 66 * 16 * Ic; idx += 256) {
            const int ii = idx & 7;
            const int nn = (idx >> 3) & 15;
            const int tl = idx >> 7;
            const int t  = t0 - 2 + tl;
            float val = (t >= 0)
                ? u[((size_t)(b0 + nn) * Tc + t) * Ic + ii] : 0.0f;
            u_lds[(tl * 16 + nn) * 9 + ii] = val;
        }
        __syncthreads();

        v8f acc = {};                             // sum over this wave's 4 channels
        float (*cbuf)[3][16] = carry[c & 1];

        for (int ii = 0; ii < 4; ++ii) {
            const int ich = ihalf * 4 + ii;
            const float* Mp = Mt + (size_t)(ich * Oc + o) * (KP * Lc);
            v8f C = {};
            #pragma unroll
            for (int kk = 0; kk < KP / 4; ++kk) {
                const int kb = kk * 4 + hi * 2;   // this lane-half's K pair
                const int m  = tile * 16 + n;     // A-operand row (time)
                v2f av, bv;
                // A 16x4 f32 layout: lanes0-15 {K0,K1}, lanes16-31 {K2,K3}
                av.x = Mp[(size_t)(kb    ) * Lc + m];
                av.y = Mp[(size_t)(kb + 1) * Lc + m];
                // B 4x16 f32 layout: lanes0-15 rows {K0,K1}, lanes16-31 {K2,K3}
                bv.x = read_x(u_lds, cbuf, kb,     n, ich);
                bv.y = read_x(u_lds, cbuf, kb + 1, n, ich);
                C = __builtin_amdgcn_wmma_f32_16x16x4_f32(
                        false, av, false, bv, (short)0, C, false, false);
            }
            acc += C;
            // carry for next chunk: rows 61/62/63 live in C VGPRs 5/6/7, hi lanes, tile 3
            if (tile == 3 && hi == 1) {
                carry[(c + 1) & 1][ich][0][n] = C[7];   // y[t0+63] -> y[-1]
                carry[(c + 1) & 1][ich][1][n] = C[6];   // y[t0+62] -> y[-2]
                carry[(c + 1) & 1][ich][2][n] = C[5];   // y[t0+61] -> y[-3]
            }
        }

        if (ihalf == 1) {
            #pragma unroll
            for (int j = 0; j < 8; ++j)
                red[tile][j + 8 * hi][n] = acc[j];
        }
        __syncthreads();
        if (ihalf == 0) {
            #pragma unroll
            for (int j = 0; j < 8; ++j) {
                const float v = acc[j] + red[tile][j + 8 * hi][n];
                const int m = tile * 16 + j + 8 * hi;     // C layout: VGPR j, lane half
                out[((size_t)(b0 + n) * Tc + (t0 + m)) * Oc + o] = v;
            }
        }
    }
}

// ---------------------------------------------------------------------------
extern "C" void kernel_launch(void* const* d_in, const int* in_sizes, int n_in,
                              void* d_out, int out_size, void* d_ws, size_t ws_size,
                              hipStream_t stream)
{
    const float* inputs      = (const float*)d_in[0];   // [B,T,I] f32
    const float* numerator   = (const float*)d_in[1];   // [I,O,3] f32
    const float* denominator = (const float*)d_in[2];   // [I,O,3] f32
    float* out = (float*)d_out;                         // [B,T,O] f32
    float* Mt  = (float*)d_ws;                          // 64 * 72 * 64 f32 = 1.18 MB

    dyno_precompute<<<1, 64, 0, stream>>>(numerator, denominator, Mt);
    dyno_main<<<Oc * (Bc / BT), 256, 0, stream>>>(inputs, Mt, out);
}